// Strict2_5DLoss_12240656793735
// MI455X (gfx1250) — compile-verified
//
#include <hip/hip_runtime.h>
#include <math.h>
#include <stdint.h>

#define ETA    3.0f
#define KCAP   96
#define POS_W  1.2f
#define NB     16
#define NG     32

typedef __attribute__((ext_vector_type(2))) float v2f;
typedef __attribute__((ext_vector_type(8))) float v8f;

__device__ __forceinline__ float softplusf(float x) {
    // log1p(exp(x)) == max(x,0) + log1p(exp(-|x|))  (stable, matches jax.nn.softplus)
    return fmaxf(x, 0.0f) + log1pf(expf(-fabsf(x)));
}

__global__ void zero_kernel(unsigned* wsU, int n) {
    int i = blockIdx.x * 256 + threadIdx.x;
    if (i < n) wsU[i] = 0u;
}

// One workgroup handles TWO (b,g) pairs (same b, g even/odd).
// Lanes 0-15 of every wave evaluate pair0's functionals, lanes 16-31 pair1's,
// sharing a single V_WMMA_F32_16X16X4_F32 per 16-point tile:
//   A rows 0..5  = {u0,u1,u2,d0,d1,d2} coefficients of triangle 0
//   A rows 8..13 = same for triangle 1
//   B cols       = 16 grid points, K features [P, Q, 1, 0]
template<int H, int W>
__global__ __launch_bounds__(256) void pair_kernel(
    const float* __restrict__ preg, const float* __restrict__ pcls,
    const float* __restrict__ gt, float* __restrict__ wsF,
    unsigned* __restrict__ wsU, float strideF, int maskOff, int pairOff)
{
    constexpr int HW = H * W;
    __shared__ float    sDist[2][HW];
    __shared__ unsigned sHist[256];
    __shared__ unsigned sCnt[256];
    __shared__ float    sGt[2][6];
    __shared__ unsigned sSel[2];
    __shared__ float    sRed[256];

    const int tid  = threadIdx.x;
    const int lane = tid & 31;
    const int wid  = tid >> 5;
    const int p0   = blockIdx.x * 2;      // pair index = b*NG + g
    const int b    = p0 / NG;
    const int g0   = p0 % NG;

    if (tid < 12) {
        int t = tid / 6, k = tid % 6;
        sGt[t][k] = gt[(size_t)((b * NG + g0 + t) * 6 + k)];
    }
    __syncthreads();

    // -------- per-lane triangle constants --------
    const int myTri = lane >> 4;
    const float Ax = sGt[myTri][0], Ay = sGt[myTri][1];
    const float Bx = sGt[myTri][2], By = sGt[myTri][3];
    const float Cx = sGt[myTri][4], Cy = sGt[myTri][5];
    const float v0x = Bx - Ax, v0y = By - Ay;       // edge A->B
    const float v1x = Cx - Bx, v1y = Cy - By;       // edge B->C
    const float v2x = Ax - Cx, v2y = Ay - Cy;       // edge C->A
    const float inv0 = 1.0f / (v0x * v0x + v0y * v0y + 1e-9f);
    const float inv1 = 1.0f / (v1x * v1x + v1y * v1y + 1e-9f);
    const float inv2 = 1.0f / (v2x * v2x + v2y * v2y + 1e-9f);

    // -------- build A fragment (16x4 f32: lane&15 = row; VGPR0: K0/K2, VGPR1: K1/K3) --------
    float cf0 = 0.0f, cf1 = 0.0f, cf2 = 0.0f;
    {
        int r  = lane & 15;
        int tA = (r >= 8) ? 1 : 0;
        int f  = r - 8 * tA;
        if (f < 6) {
            float ax = sGt[tA][0], ay = sGt[tA][1];
            float bx = sGt[tA][2], by = sGt[tA][3];
            float cx = sGt[tA][4], cy = sGt[tA][5];
            int e = (f < 3) ? f : f - 3;
            float x1, y1, x2, y2;
            if (e == 0)      { x1 = ax; y1 = ay; x2 = bx; y2 = by; }
            else if (e == 1) { x1 = bx; y1 = by; x2 = cx; y2 = cy; }
            else             { x1 = cx; y1 = cy; x2 = ax; y2 = ay; }
            if (f < 3) {
                // u = w.v = vx*P + vy*Q - (vx*x1 + vy*y1)
                float vx = x2 - x1, vy = y2 - y1;
                cf0 = vx; cf1 = vy; cf2 = -(vx * x1 + vy * y1);
            } else {
                // d = (P-x2)(y1-y2) - (x1-x2)(Q-y2) = a*P + bb*Q + c
                float a = y1 - y2, bb = x2 - x1;
                cf0 = a; cf1 = bb; cf2 = -(a * x2 + bb * y2);
            }
        }
    }
    v2f aF;
    aF.x = (lane < 16) ? cf0 : cf2;   // K0 : K2
    aF.y = (lane < 16) ? cf1 : 0.0f;  // K1 : K3

    const float INF_F = __uint_as_float(0x7F800000u);
    constexpr int NT = HW / 16;

    // -------- compute phase: distance field for both pairs --------
    for (int tile = wid; tile < NT; tile += 8) {
        int   n  = lane & 15;
        int   pt = tile * 16 + n;
        float px = ((float)(pt % W) + 0.5f) * strideF;
        float py = ((float)(pt / W) + 0.5f) * strideF;
        v2f bF;
        bF.x = (lane < 16) ? px : 1.0f;   // K0 : K2
        bF.y = (lane < 16) ? py : 0.0f;   // K1 : K3
        v8f acc = {0.f,0.f,0.f,0.f,0.f,0.f,0.f,0.f};
        acc = __builtin_amdgcn_wmma_f32_16x16x4_f32(false, aF, false, bF,
                                                    (short)0, acc, false, false);
        // lane<16: rows 0..5 (tri0); lane>=16: rows 8..13 (tri1) -> all private per lane
        float u0 = acc[0], u1 = acc[1], u2 = acc[2];
        float d0 = acc[3], d1 = acc[4], d2 = acc[5];

        float t0 = fminf(fmaxf(u0 * inv0, 0.0f), 1.0f);
        float t1 = fminf(fmaxf(u1 * inv1, 0.0f), 1.0f);
        float t2 = fminf(fmaxf(u2 * inv2, 0.0f), 1.0f);
        float ex0 = (px - Ax) - t0 * v0x, ey0 = (py - Ay) - t0 * v0y;
        float ex1 = (px - Bx) - t1 * v1x, ey1 = (py - By) - t1 * v1y;
        float ex2 = (px - Cx) - t2 * v2x, ey2 = (py - Cy) - t2 * v2y;
        float q0 = ex0 * ex0 + ey0 * ey0;
        float q1 = ex1 * ex1 + ey1 * ey1;
        float q2 = ex2 * ex2 + ey2 * ey2;
        float dist = sqrtf(fminf(q0, fminf(q1, q2)) + 1e-12f);

        float mn = fminf(d0, fminf(d1, d2));
        float mx = fmaxf(d0, fmaxf(d1, d2));
        bool inside = !((mn < 0.0f) && (mx > 0.0f));
        bool pos    = inside || (dist <= ETA);
        sDist[myTri][pt] = pos ? dist : INF_F;
    }
    __syncthreads();

    // -------- per-pair: radix-select 96 smallest, then accumulate --------
    for (int sel = 0; sel < 2; sel++) {
        unsigned prefix = 0u, maskAcc = 0u;
        unsigned kNeed = KCAP;
        for (int pass = 0; pass < 4; pass++) {
            int shift = 24 - 8 * pass;
            sHist[tid] = 0u;
            __syncthreads();
            for (int i = tid; i < HW; i += 256) {
                unsigned key = __float_as_uint(sDist[sel][i]);
                if ((key & maskAcc) == prefix)
                    atomicAdd(&sHist[(key >> shift) & 255], 1u);
            }
            __syncthreads();
            if (tid == 0) {
                unsigned cum = 0u;
                for (int j = 0; j < 256; j++) {
                    unsigned c = sHist[j];
                    if (cum + c >= kNeed) { sSel[0] = (unsigned)j; sSel[1] = kNeed - cum; break; }
                    cum += c;
                }
            }
            __syncthreads();
            prefix |= sSel[0] << shift;
            kNeed   = sSel[1];
            maskAcc |= 0xFFu << shift;
            __syncthreads();
        }
        const unsigned thr = prefix;   // key of the 96th smallest; kNeed ties to keep

        if (thr < 0x7F800000u) {
            // index-ordered tie resolution: keep first kNeed elements equal to thr
            constexpr int CH = HW / 256;
            int base = tid * CH;
            unsigned cnt = 0u;
            for (int i = 0; i < CH; i++)
                if (__float_as_uint(sDist[sel][base + i]) == thr) cnt++;
            sCnt[tid] = cnt;
            __syncthreads();
            if (tid == 0) {
                unsigned run = 0u;
                for (int j = 0; j < 256; j++) { unsigned c = sCnt[j]; sCnt[j] = run; run += c; }
            }
            __syncthreads();
            unsigned rank = sCnt[tid];
            for (int i = 0; i < CH; i++) {
                int idx = base + i;
                if (__float_as_uint(sDist[sel][idx]) == thr) {
                    if (rank >= kNeed) sDist[sel][idx] = INF_F;
                    rank++;
                }
            }
            __syncthreads();
        }

        // accumulation for pair (b, g0+sel)
        const float ax = sGt[sel][0], ay = sGt[sel][1];
        const float bx = sGt[sel][2], by = sGt[sel][3];
        const float cx = sGt[sel][4], cy = sGt[sel][5];
        const float invS = 1.0f / strideF;
        const float* pregB = preg + (size_t)b * 6 * HW;
        const float* pclsB = pcls + (size_t)b * HW;
        unsigned* maskP = wsU + maskOff + (size_t)b * (HW / 32);

        float regS = 0.0f, clsS = 0.0f; unsigned cntS = 0u;
        for (int i = tid; i < HW; i += 256) {
            unsigned key = __float_as_uint(sDist[sel][i]);
            if (key <= thr && key < 0x7F800000u) {
                cntS++;
                atomicOr(&maskP[i >> 5], 1u << (i & 31));
                clsS += softplusf(-pclsB[i]);
                float px = ((float)(i % W) + 0.5f) * strideF;
                float py = ((float)(i / W) + 0.5f) * strideF;
                float g0x = (ax - px) * invS, g0y = (ay - py) * invS;
                float g1x = (bx - px) * invS, g1y = (by - py) * invS;
                float g2x = (cx - px) * invS, g2y = (cy - py) * invS;
                float r0 = pregB[i],          r1 = pregB[HW + i];
                float r2 = pregB[2 * HW + i], r3 = pregB[3 * HW + i];
                float r4 = pregB[4 * HW + i], r5 = pregB[5 * HW + i];
                float dp0 = (r0 - g0x) * (r0 - g0x) + (r1 - g0y) * (r1 - g0y);
                float e11x = r2 - g1x, e11y = r3 - g1y;
                float e12x = r2 - g2x, e12y = r3 - g2y;
                float e21x = r4 - g1x, e21y = r5 - g1y;
                float e22x = r4 - g2x, e22y = r5 - g2y;
                float d11 = sqrtf(e11x * e11x + e11y * e11y);
                float d12 = sqrtf(e12x * e12x + e12y * e12y);
                float d21 = sqrtf(e21x * e21x + e21y * e21y);
                float d22 = sqrtf(e22x * e22x + e22y * e22y);
                float cd = fminf(d11, d12) + fminf(d21, d22)
                         + fminf(d11, d21) + fminf(d12, d22);
                regS += dp0 + cd;   // LAMBDA_P0 = LAMBDA_CD = 1
            }
        }
        float vals[3] = { regS, clsS, (float)cntS };
        for (int v = 0; v < 3; v++) {
            sRed[tid] = vals[v];
            __syncthreads();
            for (int s = 128; s > 0; s >>= 1) {
                if (tid < s) sRed[tid] += sRed[tid + s];
                __syncthreads();
            }
            if (tid == 0) wsF[pairOff + (size_t)(p0 + sel) * 4 + v] = sRed[0];
            __syncthreads();
        }
    }
}

__global__ __launch_bounds__(256) void obj_kernel(
    const float* __restrict__ pobj, const unsigned* __restrict__ wsU,
    float* __restrict__ wsF, int HW, int maskOff, int outOff)
{
    __shared__ float sL[256], sT[256];
    int tid = threadIdx.x;
    int idx = blockIdx.x * 256 + tid;       // grid exactly covers NB*HW
    int b = idx / HW, hw = idx % HW;
    unsigned bit = (wsU[maskOff + (size_t)b * (HW / 32) + (hw >> 5)] >> (hw & 31)) & 1u;
    float x = pobj[idx];
    float loss = bit ? (POS_W * softplusf(-x)) : softplusf(x);
    sL[tid] = loss; sT[tid] = (float)bit;
    __syncthreads();
    for (int s = 128; s > 0; s >>= 1) {
        if (tid < s) { sL[tid] += sL[tid + s]; sT[tid] += sT[tid + s]; }
        __syncthreads();
    }
    if (tid == 0) {
        wsF[outOff + (size_t)blockIdx.x * 2]     = sL[0];
        wsF[outOff + (size_t)blockIdx.x * 2 + 1] = sT[0];
    }
}

__global__ __launch_bounds__(256) void final_kernel(
    const float* __restrict__ wsF, float* __restrict__ out,
    int pairOff, int objOff, int nPair, int nObjPart, float totCells)
{
    __shared__ float sA[256], sB[256], sC[256], sD[256], sE[256];
    int tid = threadIdx.x;
    float reg = 0.f, cls = 0.f, pc = 0.f, ob = 0.f, ot = 0.f;
    for (int i = tid; i < nPair; i += 256) {
        const float* p = wsF + pairOff + (size_t)i * 4;
        reg += p[0]; cls += p[1]; pc += p[2];
    }
    for (int i = tid; i < nObjPart; i += 256) {
        ob += wsF[objOff + (size_t)i * 2];
        ot += wsF[objOff + (size_t)i * 2 + 1];
    }
    sA[tid] = reg; sB[tid] = cls; sC[tid] = pc; sD[tid] = ob; sE[tid] = ot;
    __syncthreads();
    for (int s = 128; s > 0; s >>= 1) {
        if (tid < s) {
            sA[tid] += sA[tid + s]; sB[tid] += sB[tid + s]; sC[tid] += sC[tid + s];
            sD[tid] += sD[tid + s]; sE[tid] += sE[tid + s];
        }
        __syncthreads();
    }
    if (tid == 0) {
        float pcs = sC[0];
        float nc = totCells - sE[0];
        float pos_eps = fmaxf(pcs, 1.0f);
        out[0] = sA[0] / pos_eps + sD[0] / (pos_eps + fmaxf(nc, 1.0f)) + sB[0] / pos_eps;
    }
}

extern "C" void kernel_launch(void* const* d_in, const int* in_sizes, int n_in,
                              void* d_out, int out_size, void* d_ws, size_t ws_size,
                              hipStream_t stream) {
    const float* preg0 = (const float*)d_in[0];
    const float* pobj0 = (const float*)d_in[1];
    const float* pcls0 = (const float*)d_in[2];
    const float* preg1 = (const float*)d_in[3];
    const float* pobj1 = (const float*)d_in[4];
    const float* pcls1 = (const float*)d_in[5];
    const float* preg2 = (const float*)d_in[6];
    const float* pobj2 = (const float*)d_in[7];
    const float* pcls2 = (const float*)d_in[8];
    const float* gtp   = (const float*)d_in[9];

    float*    wsF = (float*)d_ws;
    unsigned* wsU = (unsigned*)d_ws;

    // workspace layout (4-byte units):
    // obj masks: L0 @0 (8192 u32), L1 @8192 (2048), L2 @10240 (512)  -> 10752
    // pair partials (3 levels * 512 pairs * 4 floats) @10752          -> 16896
    // obj partials (L0:1024, L1:256, L2:64 blocks * 2 floats) @16896  -> 19584
    const int MOFF0 = 0, MOFF1 = 8192, MOFF2 = 10240;   // renamed: MASK1/MASK2 are HIP header macros
    const int PAIRO = 10752, OBJO = 16896;

    zero_kernel<<<(10752 + 255) / 256, 256, 0, stream>>>(wsU, 10752);

    pair_kernel<128, 128><<<256, 256, 0, stream>>>(preg0, pcls0, gtp, wsF, wsU, 8.0f,  MOFF0, PAIRO);
    pair_kernel< 64,  64><<<256, 256, 0, stream>>>(preg1, pcls1, gtp, wsF, wsU, 16.0f, MOFF1, PAIRO + 2048);
    pair_kernel< 32,  32><<<256, 256, 0, stream>>>(preg2, pcls2, gtp, wsF, wsU, 32.0f, MOFF2, PAIRO + 4096);

    obj_kernel<<<1024, 256, 0, stream>>>(pobj0, wsU, wsF, 16384, MOFF0, OBJO);
    obj_kernel<<< 256, 256, 0, stream>>>(pobj1, wsU, wsF, 4096,  MOFF1, OBJO + 2048);
    obj_kernel<<<  64, 256, 0, stream>>>(pobj2, wsU, wsF, 1024,  MOFF2, OBJO + 2560);

    final_kernel<<<1, 256, 0, stream>>>(wsF, (float*)d_out, PAIRO, OBJO, 1536, 1344, 344064.0f);
}